// EuclideanCodebook_89550068122197
// MI455X (gfx1250) — compile-verified
//
#include <hip/hip_runtime.h>
#include <hip/hip_bf16.h>

typedef float v2f __attribute__((ext_vector_type(2)));
typedef float v8f __attribute__((ext_vector_type(8)));

#define DECAY   0.8f
#define OMD     0.2f      // 1 - DECAY
#define EPSV    1e-5f
#define BN      16384     // B*N rows
#define DDIM    64
#define KCODES  8192
#define CH      64        // codes per LDS chunk
#define LSTR    68        // LDS row stride (floats): 16B aligned, no bank conflicts

#ifndef __has_builtin
#define __has_builtin(x) 0
#endif

// ---- CDNA5 async global->LDS (ASYNCcnt-tracked) via ISA mnemonic.
// VDST = LDS byte address (addr[31:0] of the flat LDS pointer, ISA 10.2),
// VADDR = 64-bit global address in a VGPR pair.
__device__ __forceinline__ void async_load_b128(const float* g, float* l) {
    asm volatile("global_load_async_to_lds_b128 %0, %1, off"
                 :: "v"((unsigned)(size_t)(void*)l), "v"(g) : "memory");
}

__device__ __forceinline__ void wait_async0() {
#if __has_builtin(__builtin_amdgcn_s_wait_asynccnt)
    __builtin_amdgcn_s_wait_asynccnt(0);
#else
    asm volatile("s_wait_asynccnt 0x0" ::: "memory");
#endif
}

// ---------------------------------------------------------------- e2[k] = ||E_k||^2
__global__ __launch_bounds__(256) void k_e2(const float* __restrict__ emb,
                                            float* __restrict__ e2) {
    int k = blockIdx.x * 256 + threadIdx.x;
    if (k >= KCODES) return;
    const float4* p = (const float4*)(emb + (size_t)k * DDIM);
    float s = 0.f;
#pragma unroll
    for (int i = 0; i < DDIM / 4; ++i) {
        float4 v = p[i];
        s += v.x * v.x + v.y * v.y + v.z * v.z + v.w * v.w;
    }
    e2[k] = s;
}

// ------------------------------------------------- seed new_cluster_size = DECAY*cs
__global__ __launch_bounds__(256) void k_seed_cs(const float* __restrict__ cs,
                                                 float* __restrict__ ncs) {
    int k = blockIdx.x * 256 + threadIdx.x;
    if (k < KCODES) ncs[k] = cs[k] * DECAY;
}

// --------------------------------------------------- seed new_embed_avg = DECAY*avg
__global__ __launch_bounds__(256) void k_seed_avg(const float* __restrict__ avg,
                                                  float* __restrict__ navg) {
    int g = blockIdx.x * 256 + threadIdx.x;
    if (g < KCODES * DDIM) navg[g] = avg[g] * DECAY;
}

// total = DECAY*sum(cluster_size) + OMD*BN   (counts always sum to B*N)
__global__ __launch_bounds__(256) void k_total(const float* __restrict__ cs,
                                               float* __restrict__ total) {
    __shared__ float sh[256];
    float s = 0.f;
    for (int i = threadIdx.x; i < KCODES; i += 256) s += cs[i];
    sh[threadIdx.x] = s;
    __syncthreads();
    for (int o = 128; o > 0; o >>= 1) {
        if (threadIdx.x < o) sh[threadIdx.x] += sh[threadIdx.x + o];
        __syncthreads();
    }
    if (threadIdx.x == 0) total[0] = DECAY * sh[0] + OMD * (float)BN;
}

// -------------------------------------------------------------- WMMA distance+argmin
// 256 threads (8 waves) / 128 rows of x per block; grid = 128 blocks.
// Embeddings streamed in 64-code chunks via double-buffered async global->LDS
// copies (ASYNCcnt) overlapped with 64 fp32 WMMAs per chunk. The C operand is
// seeded with -0.5*e2[col], so argmin(e2-2*cross) becomes argmax(acc) directly.
__global__ __launch_bounds__(256) void k_argmin(const float* __restrict__ x,
                                                const float* __restrict__ emb,
                                                const float* __restrict__ e2,
                                                int*   __restrict__ idx_out,
                                                float* __restrict__ ind_out) {
    __shared__ float Bsh[2][CH * LSTR];

    const int tid   = threadIdx.x;
    const int wave  = tid >> 5;
    const int lane  = tid & 31;
    const int hsel  = lane >> 4;        // 0: lanes 0-15, 1: lanes 16-31
    const int l16   = lane & 15;
    const int rowBase = blockIdx.x * 128 + wave * 16;

    // per-thread slice of the cooperative chunk fill: 4 x B128 per thread
    int lcol[4], ldv[4];
#pragma unroll
    for (int i = 0; i < 4; ++i) {
        int f  = tid + i * 256;           // float4 id in [0,1024)
        lcol[i] = f >> 4;
        ldv[i]  = (f & 15) << 2;
    }

    // A fragments: fp32 16x4 layout -> lane l16 = row; lanes 0-15 hold K={0,1},
    // lanes 16-31 hold K={2,3}.  a[kk] covers d = 4*kk + 2*hsel .. +1
    v2f a[16];
    const float* xrow = x + (size_t)(rowBase + l16) * DDIM + 2 * hsel;
#pragma unroll
    for (int kk = 0; kk < 16; ++kk)
        a[kk] = *(const v2f*)(xrow + 4 * kk);

    float bests[8];
    int   bestk[8];
#pragma unroll
    for (int v = 0; v < 8; ++v) { bests[v] = -3.4e38f; bestk[v] = 0; }

    // prologue: async-fill chunk 0 into buffer 0
#pragma unroll
    for (int i = 0; i < 4; ++i)
        async_load_b128(emb + (size_t)lcol[i] * DDIM + ldv[i],
                        &Bsh[0][lcol[i] * LSTR + ldv[i]]);

    int cur = 0;
    for (int k0 = 0; k0 < KCODES; k0 += CH) {
        wait_async0();        // my fills of Bsh[cur] have landed
        __syncthreads();      // everyone's fills done; everyone done reading Bsh[cur^1]

        if (k0 + CH < KCODES) {   // overlap next chunk's DMA with this chunk's WMMAs
#pragma unroll
            for (int i = 0; i < 4; ++i)
                async_load_b128(emb + (size_t)(k0 + CH + lcol[i]) * DDIM + ldv[i],
                                &Bsh[cur ^ 1][lcol[i] * LSTR + ldv[i]]);
        }

#pragma unroll
        for (int t = 0; t < 4; ++t) {
            const int   col  = k0 + t * 16 + l16;
            const float seed = -0.5f * e2[col];   // fold e2 into the accumulator
            v8f acc = { seed, seed, seed, seed, seed, seed, seed, seed };
            // B (4x16) fp32 layout: VGPR0 lanes0-15 K=0 / lanes16-31 K=2,
            // VGPR1 K=1/K=3 -> same d-select as A.
            const float* bs = &Bsh[cur][(t * 16 + l16) * LSTR + 2 * hsel];
#pragma unroll
            for (int kk = 0; kk < 16; ++kk) {
                v2f bv = *(const v2f*)(bs + 4 * kk);
                acc = __builtin_amdgcn_wmma_f32_16x16x4_f32(
                    /*neg_a=*/false, a[kk], /*neg_b=*/false, bv,
                    /*c_mod=*/(short)0, acc, /*reuse_a=*/false, /*reuse_b=*/false);
            }
#pragma unroll
            for (int v = 0; v < 8; ++v)     // maximize cross - 0.5*e2
                if (acc[v] > bests[v]) { bests[v] = acc[v]; bestk[v] = col; }
        }
        cur ^= 1;
    }

    // cross-lane argmax within each 16-lane half (xor masks 1..8 keep halves intact)
#pragma unroll
    for (int off = 1; off < 16; off <<= 1) {
#pragma unroll
        for (int v = 0; v < 8; ++v) {
            float os = __shfl_xor(bests[v], off, 32);
            int   ok = __shfl_xor(bestk[v], off, 32);
            if (os > bests[v] || (os == bests[v] && ok < bestk[v])) {
                bests[v] = os; bestk[v] = ok;
            }
        }
    }
    if (l16 == 0) {
        int row = rowBase + 8 * hsel;     // VGPR v holds rows v (lanes0-15) / v+8
#pragma unroll
        for (int v = 0; v < 8; ++v) {
            idx_out[row + v] = bestk[v];
            ind_out[row + v] = (float)bestk[v];
        }
    }
}

// ------------------------- gather quantize + scatter 0.2-weighted counts / x-sums
__global__ __launch_bounds__(256) void k_scatter(const float* __restrict__ x,
                                                 const float* __restrict__ emb,
                                                 const int* __restrict__ idx,
                                                 float* __restrict__ quant,
                                                 float* __restrict__ ncs,
                                                 float* __restrict__ navg) {
    int g = blockIdx.x * 256 + threadIdx.x;   // BN*DDIM threads
    if (g >= BN * DDIM) return;
    int n = g >> 6, d = g & 63;
    int code = idx[n];
    quant[g] = emb[(size_t)code * DDIM + d];
    atomicAdd(&navg[(size_t)code * DDIM + d], OMD * x[g]);
    if (d == 0) atomicAdd(&ncs[code], OMD);
}

// -------------------------------------------- Laplace-smoothed normalize of embed_avg
__global__ __launch_bounds__(256) void k_norm(const float* __restrict__ navg,
                                              const float* __restrict__ ncs,
                                              const float* __restrict__ total,
                                              float* __restrict__ out) {
    int g = blockIdx.x * 256 + threadIdx.x;   // KCODES*DDIM threads
    if (g >= KCODES * DDIM) return;
    int k = g >> 6;
    float t  = total[0];
    float c  = ncs[k];
    float sm = (c + EPSV) / (t + (float)KCODES * EPSV) * t;
    out[g] = navg[g] / sm;
}

extern "C" void kernel_launch(void* const* d_in, const int* in_sizes, int n_in,
                              void* d_out, int out_size, void* d_ws, size_t ws_size,
                              hipStream_t stream) {
    (void)in_sizes; (void)n_in; (void)out_size; (void)ws_size;

    const float* x    = (const float*)d_in[0];   // [8,2048,64]
    const float* emb  = (const float*)d_in[1];   // [1,8192,64]
    const float* cs   = (const float*)d_in[2];   // [1,8192]
    const float* avg  = (const float*)d_in[3];   // [1,8192,64]

    float* out = (float*)d_out;
    float* out_quant = out;                       // 1048576
    float* out_ind   = out + 1048576;             // 16384
    float* out_norm  = out + 1064960;             // 524288
    float* out_ncs   = out + 1589248;             // 8192
    float* out_navg  = out + 1597440;             // 524288

    float* ws     = (float*)d_ws;
    float* e2     = ws;                           // 8192
    float* total  = ws + 8192;                    // 1
    int*   idx    = (int*)(ws + 8208);            // 16384 ints

    k_e2      <<<KCODES / 256, 256, 0, stream>>>(emb, e2);
    k_seed_cs <<<KCODES / 256, 256, 0, stream>>>(cs, out_ncs);
    k_seed_avg<<<(KCODES * DDIM) / 256, 256, 0, stream>>>(avg, out_navg);
    k_total   <<<1, 256, 0, stream>>>(cs, total);
    k_argmin  <<<BN / 128, 256, 0, stream>>>(x, emb, e2, idx, out_ind);
    k_scatter <<<(BN * DDIM) / 256, 256, 0, stream>>>(x, emb, idx,
                                                      out_quant, out_ncs, out_navg);
    k_norm    <<<(KCODES * DDIM) / 256, 256, 0, stream>>>(out_navg, out_ncs,
                                                          total, out_norm);
}